// KVGather_23785528885338
// MI455X (gfx1250) — compile-verified
//
#include <hip/hip_runtime.h>

// ---------------------------------------------------------------------------
// KV top-k region gather for MI455X (gfx1250).
//
// r_idx (4,64,16) int32, kv (4,64,49,512) f32 -> out (4,64,16,49,512) f32.
// 4096 independent contiguous copies of 100,352 B. Zero FLOPs -> HBM-write
// bound: 411 MB out + 25.7 MB unique reads ~= 19 us at 23.3 TB/s. kv is
// re-read 16x and fits the 192 MB L2 (keep loads RT); output is streamed
// with non-temporal stores.
//
// Launched path: CDNA5 async copy engine (ASYNCcnt), global->LDS->global
// b128, double-buffered 25 KB quarters with s_wait_asynccnt 0 fences
// (loads-vs-stores complete out of order, so 0 is the only safe cross-type
// fence). Verified lowering in round 2.
//
// Probe path (compiled, not launched yet): Tensor Data Mover via
// __builtin_amdgcn_tensor_load_to_lds / _store_from_lds. TDM ops are
// in-order with each other on TENSORcnt, so threshold waits pipeline two
// 50 KB halves per block with just 4 instructions + 3 waits.
// ---------------------------------------------------------------------------

constexpr int kB    = 4;
constexpr int kP2   = 64;
constexpr int kTopK = 16;
constexpr int kW2   = 49;
constexpr int kCkv  = 512;

constexpr int kRegionFloats = kW2 * kCkv;         // 25088 floats = 100,352 B
constexpr int kRegionVec4   = kRegionFloats / 4;  // 6272 b128 chunks
constexpr int kNumRegions   = kB * kP2 * kTopK;   // 4096
constexpr int kThreads      = 256;                // 8 wave32s per block
constexpr int kQuarterVec4  = kRegionVec4 / 4;    // 1568 b128 = 25,088 B

typedef float v4f __attribute__((ext_vector_type(4)));

// -------------------------- async (ASYNCcnt) path ---------------------------

__device__ __forceinline__ void async_wait0() {
  asm volatile("s_wait_asynccnt 0x0" ::: "memory");
}

__device__ __forceinline__ void async_load_b128(unsigned int lds_off,
                                                unsigned long long gaddr) {
  asm volatile("global_load_async_to_lds_b128 %0, %1, off"
               :: "v"(lds_off), "v"(gaddr)
               : "memory");
}

__device__ __forceinline__ void async_store_b128_nt(unsigned long long gaddr,
                                                    unsigned int lds_off) {
  asm volatile("global_store_async_from_lds_b128 %0, %1, off th:TH_STORE_NT"
               :: "v"(gaddr), "v"(lds_off)
               : "memory");
}

__global__ __launch_bounds__(kThreads)
void KVGather_async_kernel(const int* __restrict__ r_idx,
                           const float* __restrict__ kv,
                           float* __restrict__ out) {
  __shared__ v4f lds[2][kQuarterVec4];  // 50,176 B -> 6 blocks / 320 KB WGP

  const int region = blockIdx.x;        // flat (b,p,k) == r_idx flat layout
  const int b      = region >> 10;
  const int idx    = r_idx[region];     // uniform -> scalar load

  const v4f* __restrict__ src =
      reinterpret_cast<const v4f*>(kv) + (size_t)(b * kP2 + idx) * kRegionVec4;
  v4f* __restrict__ dst =
      reinterpret_cast<v4f*>(out) + (size_t)region * kRegionVec4;

  const int tid = threadIdx.x;
  __builtin_prefetch(src + tid, 0, 1);  // global_prefetch_b8: warm L2

  auto issue_loads = [&](int q) {
    const int buf = q & 1;
    for (int i = tid; i < kQuarterVec4; i += kThreads) {
      unsigned long long ga =
          (unsigned long long)(src + (size_t)q * kQuarterVec4 + i);
      unsigned int la = (unsigned int)(unsigned long long)&lds[buf][i];
      async_load_b128(la, ga);
    }
  };
  auto issue_stores = [&](int q) {
    const int buf = q & 1;
    for (int i = tid; i < kQuarterVec4; i += kThreads) {
      unsigned long long ga =
          (unsigned long long)(dst + (size_t)q * kQuarterVec4 + i);
      unsigned int la = (unsigned int)(unsigned long long)&lds[buf][i];
      async_store_b128_nt(ga, la);
    }
  };

  issue_loads(0);
  for (int q = 1; q < 4; ++q) {
    async_wait0();       // drains L(q-1) (S(q-1) may read LDS) and S(q-2)
    issue_stores(q - 1); // (L(q) may overwrite buffer q&1)
    issue_loads(q);
  }
  async_wait0();
  issue_stores(3);
  async_wait0();
}

// ------------------------ TDM (TENSORcnt) probe path ------------------------
// One wave per block; the TDM DMA engine moves each 50,176 B half-region with
// a single instruction. D# bitfields per CDNA5 ISA ch.8:
//   group0: [1:0]=count=1, [63:32]=lds_addr, [120:64]=global_addr,
//           [127:126]=type=2
//   group1: [17:16]=data_size (3 = 8 B elems), [79:48]=tensor_dim0,
//           [111:80]=tensor_dim1, [127:112]=tile_dim0, [143:128]=tile_dim1,
//           [207:160]=tensor_dim0_stride, [255:208]=tensor_dim1_stride
//   groups 2/3: zero (tile_dim3/4 = 0 -> unused; 1-D copy)

#if __has_builtin(__builtin_amdgcn_tensor_load_to_lds) && \
    __has_builtin(__builtin_amdgcn_tensor_store_from_lds) && \
    __has_builtin(__builtin_amdgcn_s_wait_tensorcnt)
#define HAVE_TDM 1
#endif

#ifdef HAVE_TDM
typedef unsigned int u32x4 __attribute__((ext_vector_type(4)));
typedef int          i32x4 __attribute__((ext_vector_type(4)));
typedef int          i32x8 __attribute__((ext_vector_type(8)));

constexpr unsigned kHalfElems8 = kRegionFloats / 2 / 2;  // 6272 8-byte elems

__device__ __forceinline__ u32x4 tdm_group0(unsigned lds_addr,
                                            unsigned long long gaddr) {
  u32x4 g;
  g.x = 1u;                                        // count=1, user descriptor
  g.y = lds_addr;                                  // LDS byte address
  g.z = (unsigned)(gaddr & 0xFFFFFFFFull);         // global_addr[31:0]
  g.w = ((unsigned)(gaddr >> 32) & 0x01FFFFFFu)    // global_addr[56:32]
        | (2u << 30);                              // type=2 ("image")
  return g;
}

__device__ __forceinline__ i32x8 tdm_group1() {
  i32x8 g;
  g[0] = (int)(3u << 16);                          // data_size=3 (8 B)
  g[1] = (int)((kHalfElems8 & 0xFFFFu) << 16);     // tensor_dim0[15:0]
  g[2] = (int)(((kHalfElems8 >> 16) & 0xFFFFu) |   // tensor_dim0[31:16]
               (1u << 16));                        // tensor_dim1 = 1
  g[3] = (int)((kHalfElems8 & 0xFFFFu) << 16);     // tile_dim0 = 6272
  g[4] = 1;                                        // tile_dim1 = 1
  g[5] = (int)kHalfElems8;                         // tensor_dim0_stride[31:0]
  g[6] = (int)((kHalfElems8 & 0xFFFFu) << 16);     // tensor_dim1_stride[15:0]
  g[7] = 0;
  return g;
}

__device__ __forceinline__ void tdm_load(u32x4 g0, i32x8 g1) {
  i32x4 z4 = {0, 0, 0, 0};
#if __has_include(<hip/amd_detail/amd_gfx1250_TDM.h>)
  i32x8 z8 = {0, 0, 0, 0, 0, 0, 0, 0};             // clang-23 6-arg form
  __builtin_amdgcn_tensor_load_to_lds(g0, g1, z4, z4, z8, 0);
#else
  __builtin_amdgcn_tensor_load_to_lds(g0, g1, z4, z4, 0);  // ROCm 7.2 5-arg
#endif
}

__device__ __forceinline__ void tdm_store(u32x4 g0, i32x8 g1) {
  i32x4 z4 = {0, 0, 0, 0};
#if __has_include(<hip/amd_detail/amd_gfx1250_TDM.h>)
  i32x8 z8 = {0, 0, 0, 0, 0, 0, 0, 0};
  __builtin_amdgcn_tensor_store_from_lds(g0, g1, z4, z4, z8, 0);
#else
  __builtin_amdgcn_tensor_store_from_lds(g0, g1, z4, z4, 0);
#endif
}

__global__ __launch_bounds__(32)
void KVGather_tdm_kernel(const int* __restrict__ r_idx,
                         const float* __restrict__ kv,
                         float* __restrict__ out) {
  __shared__ v4f lds[2][kRegionVec4 / 2];  // 2 x 50,176 B = full region

  const int region = blockIdx.x;
  const int b      = region >> 10;
  const int idx    = r_idx[region];

  const unsigned long long src =
      (unsigned long long)(kv) +
      (unsigned long long)(b * kP2 + idx) * (kRegionFloats * 4ull);
  const unsigned long long dst =
      (unsigned long long)(out) +
      (unsigned long long)region * (kRegionFloats * 4ull);
  constexpr unsigned long long kHalfBytes = kRegionFloats * 2ull;  // 50,176 B

  const unsigned lds0 = (unsigned)(unsigned long long)&lds[0][0];
  const unsigned lds1 = (unsigned)(unsigned long long)&lds[1][0];
  const i32x8 g1 = tdm_group1();

  // In-order TENSORcnt pipeline: TL0, TL1, (<=1 -> TL0 done), TS0,
  // (<=1 -> TL1 done), TS1, (0 -> all stores complete).
  tdm_load(tdm_group0(lds0, src), g1);
  tdm_load(tdm_group0(lds1, src + kHalfBytes), g1);
  __builtin_amdgcn_s_wait_tensorcnt(1);
  tdm_store(tdm_group0(lds0, dst), g1);
  __builtin_amdgcn_s_wait_tensorcnt(1);
  tdm_store(tdm_group0(lds1, dst + kHalfBytes), g1);
  __builtin_amdgcn_s_wait_tensorcnt(0);
}
#endif  // HAVE_TDM

// ------------------ direct VGPR copy path (known-good fallback) -------------

constexpr int kFullIters = kRegionVec4 / kThreads;  // 24 (+128-lane tail)

__global__ __launch_bounds__(kThreads)
void KVGather_direct_kernel(const int* __restrict__ r_idx,
                            const float* __restrict__ kv,
                            float* __restrict__ out) {
  const int region = blockIdx.x;
  const int b      = region >> 10;
  const int idx    = r_idx[region];

  const v4f* __restrict__ src =
      reinterpret_cast<const v4f*>(kv) + (size_t)(b * kP2 + idx) * kRegionVec4;
  v4f* __restrict__ dst =
      reinterpret_cast<v4f*>(out) + (size_t)region * kRegionVec4;

  const int tid = threadIdx.x;
  __builtin_prefetch(src + tid, 0, 1);

  int i = tid;
#pragma unroll 8
  for (int j = 0; j < kFullIters; ++j, i += kThreads) {
    v4f v = src[i];
    __builtin_nontemporal_store(v, dst + i);
  }
  if (i < kRegionVec4) {
    v4f v = src[i];
    __builtin_nontemporal_store(v, dst + i);
  }
}

// ----------------------------------------------------------------------------

extern "C" void kernel_launch(void* const* d_in, const int* in_sizes, int n_in,
                              void* d_out, int out_size, void* d_ws, size_t ws_size,
                              hipStream_t stream) {
  const int*   r_idx = (const int*)d_in[0];
  const float* kv    = (const float*)d_in[1];
  float*       out   = (float*)d_out;

  (void)in_sizes; (void)n_in; (void)out_size; (void)d_ws; (void)ws_size;

  KVGather_async_kernel<<<kNumRegions, kThreads, 0, stream>>>(r_idx, kv, out);
}